// GNN_Graph_31653908971943
// MI455X (gfx1250) — compile-verified
//
#include <hip/hip_runtime.h>
#include <hip/hip_bf16.h>
#include <math.h>

// ---------------------------------------------------------------------------
// GINE-style GNN, 5 layers, fp32 throughout.
// Matrix ops use V_WMMA_F32_16X16X4_F32 (fp32 WMMA, wave32, 16x16 tiles).
// ---------------------------------------------------------------------------

typedef float v2f __attribute__((ext_vector_type(2)));
typedef float v8f __attribute__((ext_vector_type(8)));

static constexpr int N_NODES = 100000;
static constexpr int N_EDGES = 400000;
static constexpr int N_GRAPH = 4096;
static constexpr int C_IN    = 64;
static constexpr int C_HID   = 128;   // all layers after the first
static constexpr int C_ED    = 16;
static constexpr int N_LAY   = 5;
static constexpr float BN_EPS = 1e-5f;

__device__ __forceinline__ v8f wmma_f32_4(v2f a, v2f b, v8f c) {
  // D = A(16x4,f32) * B(4x16,f32) + C(16x16,f32)
  return __builtin_amdgcn_wmma_f32_16x16x4_f32(
      /*neg_a=*/false, a, /*neg_b=*/false, b,
      /*c_mod=*/(short)0, c, /*reuse_a=*/false, /*reuse_b=*/false);
}

// ---------------------------------------------------------------------------
// zero fill
// ---------------------------------------------------------------------------
__global__ void zero_kernel(float* __restrict__ p, int n) {
  int i = blockIdx.x * blockDim.x + threadIdx.x;
  if (i < n) p[i] = 0.0f;
}

// ---------------------------------------------------------------------------
// Fused edge MLP + message + scatter-add:
//   agg[dst] += relu( x[src] + edge_attr @ We + be )
// One wave == 16 edges. K = ED = 16 -> 4 WMMA(16x16x4) steps per col tile.
// ---------------------------------------------------------------------------
__global__ void edge_msg_kernel(const float* __restrict__ x,
                                const float* __restrict__ ea,
                                const float* __restrict__ We,   // [ED, cin]
                                const float* __restrict__ be,   // [cin]
                                const int*   __restrict__ src,
                                const int*   __restrict__ dst,
                                float* __restrict__ agg,        // [N, cin]
                                int nEdges, int cin) {
  const int wave = (blockIdx.x * blockDim.x + threadIdx.x) >> 5;
  const int lane = threadIdx.x & 31;
  const int ntiles = nEdges >> 4;          // E divisible by 16
  if (wave >= ntiles) return;              // wave-uniform exit: EXEC stays full

  const int e0   = wave << 4;
  const int half = lane >> 4;              // 0: K pair {k,k+1}, 1: {k+2,k+3}
  const int l16  = lane & 15;
  const int ctiles = cin >> 4;

  for (int ct = 0; ct < ctiles; ++ct) {
    const int c0 = ct << 4;
    v8f acc = {0.f, 0.f, 0.f, 0.f, 0.f, 0.f, 0.f, 0.f};
#pragma unroll
    for (int k = 0; k < C_ED; k += 4) {
      const int kk = k + 2 * half;
      v2f a, b;
      a[0] = ea[(e0 + l16) * C_ED + kk];
      a[1] = ea[(e0 + l16) * C_ED + kk + 1];
      b[0] = We[(kk    ) * cin + c0 + l16];
      b[1] = We[(kk + 1) * cin + c0 + l16];
      acc = wmma_f32_4(a, b, acc);
    }
    const int   col  = c0 + l16;
    const float bias = be[col];
#pragma unroll
    for (int r = 0; r < 8; ++r) {
      const int e = e0 + r + 8 * half;     // D layout: row = vgpr + 8*(lane>=16)
      const int s = src[e];
      const int d = dst[e];
      float v = acc[r] + bias + x[(long)s * cin + col];
      v = fmaxf(v, 0.0f);
      atomicAdd(&agg[(long)d * cin + col], v);
    }
  }
}

// ---------------------------------------------------------------------------
// C[M,NC] = (A1 (+ A2)) [M,K] @ W[K,NC] + bias[NC]
// One 16x16 output tile per wave; K stepped by 4 with fp32 WMMA.
// ---------------------------------------------------------------------------
__global__ void gemm_wmma_kernel(const float* __restrict__ A1,
                                 const float* __restrict__ A2,  // nullable
                                 const float* __restrict__ W,
                                 const float* __restrict__ bias,
                                 float* __restrict__ C,
                                 int M, int K, int NC) {
  const int wave = (blockIdx.x * blockDim.x + threadIdx.x) >> 5;
  const int lane = threadIdx.x & 31;
  const int ctiles = NC >> 4;
  const int ntiles = (M >> 4) * ctiles;    // M divisible by 16
  if (wave >= ntiles) return;              // wave-uniform

  const int r0   = (wave / ctiles) << 4;
  const int c0   = (wave % ctiles) << 4;
  const int half = lane >> 4;
  const int l16  = lane & 15;
  const long arow = r0 + l16;

  v8f acc = {0.f, 0.f, 0.f, 0.f, 0.f, 0.f, 0.f, 0.f};
  for (int k = 0; k < K; k += 4) {
    const int kk = k + 2 * half;
    v2f a, b;
    a[0] = A1[arow * K + kk];
    a[1] = A1[arow * K + kk + 1];
    if (A2) {                              // uniform (SGPR) branch
      a[0] += A2[arow * K + kk];
      a[1] += A2[arow * K + kk + 1];
    }
    b[0] = W[(kk    ) * NC + c0 + l16];
    b[1] = W[(kk + 1) * NC + c0 + l16];
    acc = wmma_f32_4(a, b, acc);
  }
  const int   col = c0 + l16;
  const float bv  = bias[col];
#pragma unroll
  for (int r = 0; r < 8; ++r) {
    const long row = r0 + r + 8 * half;
    C[row * NC + col] = acc[r] + bv;
  }
}

// ---------------------------------------------------------------------------
// BatchNorm stats: per-feature sum and sum-of-squares over N rows.
// blockDim == 128 (== C_HID); thread t owns column t. Coalesced row reads,
// 2 atomics per thread per block into 256 hot addresses.
// ---------------------------------------------------------------------------
__global__ void bn_stats_kernel(const float* __restrict__ h,
                                float* __restrict__ stats,   // [2*128]
                                int nRows) {
  const int c = threadIdx.x;               // 0..127
  float s = 0.0f, s2 = 0.0f;
  for (int r = blockIdx.x; r < nRows; r += gridDim.x) {
    const float v = h[(long)r * C_HID + c];
    s  += v;
    s2 += v * v;
  }
  atomicAdd(&stats[c], s);
  atomicAdd(&stats[C_HID + c], s2);
}

// ---------------------------------------------------------------------------
// Fused BatchNorm (batch stats) + ELU:
//   x = elu( (h - mu) * rsqrt(var + eps) * gamma + beta )
// ---------------------------------------------------------------------------
__global__ void bn_elu_kernel(const float* __restrict__ h,
                              const float* __restrict__ stats,
                              const float* __restrict__ gamma,
                              const float* __restrict__ beta,
                              float* __restrict__ xo,
                              int nElem) {
  const int i = blockIdx.x * blockDim.x + threadIdx.x;
  if (i >= nElem) return;
  const int c = i & (C_HID - 1);
  const float invN = 1.0f / (float)N_NODES;
  const float mu  = stats[c] * invN;
  const float var = stats[C_HID + c] * invN - mu * mu;   // biased var (jnp.var)
  const float scale = gamma[c] * rsqrtf(var + BN_EPS);
  float v = (h[i] - mu) * scale + beta[c];
  v = (v > 0.0f) ? v : (expf(v) - 1.0f);                 // ELU (alpha=1)
  xo[i] = v;
}

// ---------------------------------------------------------------------------
// global_add_pool: pooled[batch[n], c] += x[n, c]
// ---------------------------------------------------------------------------
__global__ void pool_kernel(const float* __restrict__ x,
                            const int* __restrict__ batch,
                            float* __restrict__ pooled,
                            int nElem) {
  const int i = blockIdx.x * blockDim.x + threadIdx.x;
  if (i >= nElem) return;
  const int n = i >> 7;                    // /128
  const int c = i & (C_HID - 1);
  atomicAdd(&pooled[(long)batch[n] * C_HID + c], x[i]);
}

// ---------------------------------------------------------------------------
// host launch
// ---------------------------------------------------------------------------
extern "C" void kernel_launch(void* const* d_in, const int* in_sizes, int n_in,
                              void* d_out, int out_size, void* d_ws, size_t ws_size,
                              hipStream_t stream) {
  (void)in_sizes; (void)n_in; (void)out_size; (void)ws_size;

  // inputs in setup_inputs() dict order (lists flattened element-by-element)
  const float* x0    = (const float*)d_in[0];
  const int*   ei    = (const int*)  d_in[1];   // [2, E]
  const float* ea    = (const float*)d_in[2];   // [E, 16]
  const int*   batch = (const int*)  d_in[3];   // [N]
  const float* We[N_LAY], *be[N_LAY], *Wn[N_LAY], *bn[N_LAY], *ga[N_LAY], *bt[N_LAY];
  for (int l = 0; l < N_LAY; ++l) {
    We[l] = (const float*)d_in[4  + l];
    be[l] = (const float*)d_in[9  + l];
    Wn[l] = (const float*)d_in[14 + l];
    bn[l] = (const float*)d_in[19 + l];
    ga[l] = (const float*)d_in[24 + l];
    bt[l] = (const float*)d_in[29 + l];
  }
  const float* Wlin = (const float*)d_in[34];
  const float* blin = (const float*)d_in[35];

  const int* src = ei;
  const int* dst = ei + N_EDGES;

  // workspace carve-up (fp32): xbuf | agg | hbuf | stats  (~154 MB)
  float* ws    = (float*)d_ws;
  float* xbuf  = ws;
  float* agg   = xbuf + (long)N_NODES * C_HID;
  float* hbuf  = agg  + (long)N_NODES * C_HID;
  float* stats = hbuf + (long)N_NODES * C_HID;   // 256 floats
  float* pooled = agg;                           // agg is dead after last layer

  const float* xcur = x0;
  int cin = C_IN;

  for (int l = 0; l < N_LAY; ++l) {
    const int cout = C_HID;

    // agg = 0
    {
      const int n = N_NODES * cin;
      zero_kernel<<<(n + 255) / 256, 256, 0, stream>>>(agg, n);
    }
    // agg[dst] += relu(x[src] + ea @ We + be)
    {
      const int waves  = N_EDGES / 16;
      const int blocks = (waves + 7) / 8;        // 8 waves / 256-thread block
      edge_msg_kernel<<<blocks, 256, 0, stream>>>(xcur, ea, We[l], be[l],
                                                  src, dst, agg, N_EDGES, cin);
    }
    // h = (x + agg) @ Wn + bn
    {
      const int tiles  = (N_NODES / 16) * (cout / 16);
      const int blocks = (tiles + 7) / 8;
      gemm_wmma_kernel<<<blocks, 256, 0, stream>>>(xcur, agg, Wn[l], bn[l],
                                                   hbuf, N_NODES, cin, cout);
    }
    // batch-norm stats + fused BN/ELU
    zero_kernel<<<1, 256, 0, stream>>>(stats, 2 * C_HID);
    bn_stats_kernel<<<512, C_HID, 0, stream>>>(hbuf, stats, N_NODES);
    {
      const int n = N_NODES * cout;
      bn_elu_kernel<<<(n + 255) / 256, 256, 0, stream>>>(hbuf, stats, ga[l],
                                                         bt[l], xbuf, n);
    }
    xcur = xbuf;
    cin  = C_HID;
  }

  // global add pool
  {
    const int n = N_GRAPH * C_HID;
    zero_kernel<<<(n + 255) / 256, 256, 0, stream>>>(pooled, n);
  }
  {
    const int n = N_NODES * C_HID;
    pool_kernel<<<(n + 255) / 256, 256, 0, stream>>>(xbuf, batch, pooled, n);
  }
  // out = pooled @ Wlin + blin
  {
    const int tiles  = (N_GRAPH / 16) * (C_HID / 16);
    const int blocks = (tiles + 7) / 8;
    gemm_wmma_kernel<<<blocks, 256, 0, stream>>>(pooled, nullptr, Wlin, blin,
                                                 (float*)d_out, N_GRAPH,
                                                 C_HID, C_HID);
  }
}